// DeformableAttention_25786983646079
// MI455X (gfx1250) — compile-verified
//
#include <hip/hip_runtime.h>
#include <cstdint>

typedef __attribute__((ext_vector_type(2))) float v2f;
typedef __attribute__((ext_vector_type(8))) float v8f;

#define BQ   8
#define NQ   4096
#define DM   256
#define NHH  8
#define NPP  4
#define DH   32
#define HS_  64
#define WS_  64

// ---------------------------------------------------------------------------
// C[M, NCOLS] = A[M, K] * W[K, NCOLS] + bias   (all row-major fp32)
// Block: 256 threads = 8 wave32; each wave -> one 16x16 tile via
// v_wmma_f32_16x16x4_f32. Weight slab (K x 16) staged in LDS per block.
// grid = (M/128, NCOLS/16)
// ---------------------------------------------------------------------------
template <int K, int NCOLS>
__launch_bounds__(256)
__global__ void wmma_gemm_bias(const float* __restrict__ A,
                               const float* __restrict__ Wm,
                               const float* __restrict__ bias,
                               float* __restrict__ C)
{
    __shared__ float lds_w[K * 16];

    const int tid     = threadIdx.x;
    const int colBase = blockIdx.y * 16;

    // Stage W[:, colBase:colBase+16] into LDS (coalesced across threads).
    for (int e = tid; e < K * 16; e += 256) {
        const int k = e >> 4;
        const int n = e & 15;
        lds_w[e] = Wm[k * NCOLS + colBase + n];
    }
    __syncthreads();

    const int lane = tid & 31;
    const int wave = tid >> 5;
    const int hi   = lane >> 4;   // lane half selects K-pair (A/B) / M-half (C)
    const int lo   = lane & 15;   // M index for A, N index for B/C

    const int rowBase = blockIdx.x * 128 + wave * 16;
    const float* __restrict__ arow = A + (size_t)(rowBase + lo) * K;
    __builtin_prefetch(arow, 0, 1);   // global_prefetch_b8

    // Init accumulator with bias broadcast along columns.
    v8f c;
    const float bv = bias[colBase + lo];
#pragma unroll
    for (int r = 0; r < 8; ++r) c[r] = bv;

#pragma unroll 8
    for (int k0 = 0; k0 < K; k0 += 4) {
        const int ka = k0 + 2 * hi;
        // A 16x4 layout: lane(lo) = row M, VGPR j holds K = k0 + 2*hi + j
        v2f a;
        a.x = arow[ka];
        a.y = arow[ka + 1];
        // B 4x16 layout: lane(lo) = col N, VGPR j holds K = k0 + 2*hi + j
        v2f b;
        b.x = lds_w[ka * 16 + lo];
        b.y = lds_w[(ka + 1) * 16 + lo];
        c = __builtin_amdgcn_wmma_f32_16x16x4_f32(
                /*neg_a=*/false, a, /*neg_b=*/false, b,
                /*c_mod=*/(short)0, c, /*reuse_a=*/false, /*reuse_b=*/false);
    }

    // C/D layout: VGPR r -> row M = r + 8*hi, col N = lo
    const int mBase = rowBase + 8 * hi;
#pragma unroll
    for (int r = 0; r < 8; ++r) {
        C[(size_t)(mBase + r) * NCOLS + colBase + lo] = c[r];
    }
}

// ---------------------------------------------------------------------------
// Deformable sampling: one wave32 per (b, n, h); lane = channel within head.
// value:  [B*N, NH*DH]   (projected, head-major channels)
// refpts: [B*N, 2]
// offs:   [B*N, NH*NP*2] (raw offset logits)
// attn:   [B*N, NH*NP]   (softmax logits)
// out:    [B*N, NH*DH]
// ---------------------------------------------------------------------------
__launch_bounds__(256)
__global__ void deform_sample_kernel(const float* __restrict__ value,
                                     const float* __restrict__ refpts,
                                     const float* __restrict__ offs,
                                     const float* __restrict__ attn,
                                     float* __restrict__ out)
{
    const int lane  = threadIdx.x & 31;             // channel c in [0, DH)
    const int gwave = blockIdx.x * 8 + (threadIdx.x >> 5);
    const int h     = gwave & (NHH - 1);
    const int bn    = gwave >> 3;                   // b*NQ + n
    const int b     = bn >> 12;                     // NQ == 4096

    const float rx = refpts[bn * 2 + 0];
    const float ry = refpts[bn * 2 + 1];

    // softmax over the NP sampling points of this head (wave-uniform)
    float lg[NPP];
    float mx = -1e30f;
#pragma unroll
    for (int p = 0; p < NPP; ++p) {
        lg[p] = attn[(size_t)bn * (NHH * NPP) + h * NPP + p];
        mx = fmaxf(mx, lg[p]);
    }
    float ssum = 0.f;
#pragma unroll
    for (int p = 0; p < NPP; ++p) { lg[p] = __expf(lg[p] - mx); ssum += lg[p]; }
    const float inv = 1.f / ssum;

    float acc = 0.f;
#pragma unroll
    for (int p = 0; p < NPP; ++p) {
        const size_t ob = (size_t)bn * (NHH * NPP * 2) + (h * NPP + p) * 2;
        const float ox = offs[ob + 0];
        const float oy = offs[ob + 1];
        float lx = rx + ox * (1.0f / WS_);
        float ly = ry + oy * (1.0f / HS_);
        lx = fminf(fmaxf(lx, 0.f), 1.f);
        ly = fminf(fmaxf(ly, 0.f), 1.f);
        // grid = loc*2-1; ix = ((gx+1)*W - 1)*0.5  ==  loc*W - 0.5
        const float ix  = lx * WS_ - 0.5f;
        const float iy  = ly * HS_ - 0.5f;
        const float x0f = floorf(ix);
        const float y0f = floorf(iy);
        const float wx1 = ix - x0f, wy1 = iy - y0f;
        const float wx0 = 1.f - wx1, wy0 = 1.f - wy1;
        const int   x0  = (int)x0f, y0 = (int)y0f;
        const float wp  = lg[p] * inv;

        float sval = 0.f;
#pragma unroll
        for (int cy = 0; cy < 2; ++cy) {
#pragma unroll
            for (int cx = 0; cx < 2; ++cx) {
                const int xi = x0 + cx;
                const int yi = y0 + cy;
                const float cw = (cx ? wx1 : wx0) * (cy ? wy1 : wy0);
                if (xi >= 0 && xi < WS_ && yi >= 0 && yi < HS_) {  // wave-uniform
                    const size_t vrow = (size_t)(b * NQ + yi * WS_ + xi);
                    sval += cw * value[vrow * DM + h * DH + lane];
                }
            }
        }
        acc += wp * sval;
    }
    out[(size_t)bn * DM + h * DH + lane] = acc;
}

// ---------------------------------------------------------------------------
extern "C" void kernel_launch(void* const* d_in, const int* in_sizes, int n_in,
                              void* d_out, int out_size, void* d_ws, size_t ws_size,
                              hipStream_t stream)
{
    const float* query  = (const float*)d_in[0];
    const float* refpts = (const float*)d_in[1];
    const float* vsrc   = (const float*)d_in[2];
    // d_in[3], d_in[4]: spatial_h/w scalars (compile-time constants here)
    const float* W_off  = (const float*)d_in[5];
    const float* b_off  = (const float*)d_in[6];
    const float* W_attn = (const float*)d_in[7];
    const float* b_attn = (const float*)d_in[8];
    const float* W_val  = (const float*)d_in[9];
    const float* b_val  = (const float*)d_in[10];
    const float* W_out  = (const float*)d_in[11];
    const float* b_out  = (const float*)d_in[12];
    float* out = (float*)d_out;

    const int M = BQ * NQ;  // 32768

    char* ws = (char*)d_ws;
    float* ws_value = (float*)ws; ws += (size_t)M * DM * sizeof(float);
    float* ws_off   = (float*)ws; ws += (size_t)M * (NHH * NPP * 2) * sizeof(float);
    float* ws_attn  = (float*)ws; ws += (size_t)M * (NHH * NPP) * sizeof(float);
    float* ws_pre   = (float*)ws;

    dim3 blk(256);
    // value projection:  [M,256] x [256,256]
    wmma_gemm_bias<DM, DM><<<dim3(M / 128, DM / 16), blk, 0, stream>>>(
        vsrc, W_val, b_val, ws_value);
    // offset logits:     [M,256] x [256,64]
    wmma_gemm_bias<DM, 64><<<dim3(M / 128, 4), blk, 0, stream>>>(
        query, W_off, b_off, ws_off);
    // attention logits:  [M,256] x [256,32]
    wmma_gemm_bias<DM, 32><<<dim3(M / 128, 2), blk, 0, stream>>>(
        query, W_attn, b_attn, ws_attn);
    // bilinear sampling + softmax + per-head weighted sum
    deform_sample_kernel<<<dim3(M * NHH / 8), blk, 0, stream>>>(
        ws_value, refpts, ws_off, ws_attn, ws_pre);
    // output projection: [M,256] x [256,256]
    wmma_gemm_bias<DM, DM><<<dim3(M / 128, DM / 16), blk, 0, stream>>>(
        ws_pre, W_out, b_out, out);
}